// GNNModel_76656576299690
// MI455X (gfx1250) — compile-verified
//
#include <hip/hip_runtime.h>
#include <hip/hip_bf16.h>

typedef __attribute__((ext_vector_type(2))) float v2f;
typedef __attribute__((ext_vector_type(8))) float v8f;

#define N_NODES_C 100000

// ---------------------------------------------------------------------------
// Utility: zero-fill
// ---------------------------------------------------------------------------
__global__ void zero_f32(float* __restrict__ p, int n) {
    int i = blockIdx.x * blockDim.x + threadIdx.x;
    if (i < n) p[i] = 0.0f;
}

// ---------------------------------------------------------------------------
// Degree count: deg[dst] += 1 over edges  (deg buffer pre-zeroed)
// ---------------------------------------------------------------------------
__global__ void deg_count(const int* __restrict__ dst, float* __restrict__ deg, int ne) {
    int e = blockIdx.x * blockDim.x + threadIdx.x;
    if (e < ne) unsafeAtomicAdd(&deg[dst[e]], 1.0f);
}

// dis[i] = rsqrt(deg[i] + 1)   (in place)
__global__ void finish_dis(float* __restrict__ deg, int n) {
    int i = blockIdx.x * blockDim.x + threadIdx.x;
    if (i < n) deg[i] = rsqrtf(deg[i] + 1.0f);
}

// ---------------------------------------------------------------------------
// WMMA fp32 GEMM:  Y[M x N] = X[M x K] * W[K x N]   (optional bias + relu)
// One wave computes a 16-row block across all N columns (NT tiles of 16).
// K iterated in steps of 4 using V_WMMA_F32_16X16X4_F32.
//
// A (16x4 f32) per-lane layout (ISA 7.12.2): lane&15 = M row,
//   VGPR0 = K = 2*(lane>>4), VGPR1 = K+1.
// B (4x16 f32): lane&15 = N col, VGPR0 = K = 2*(lane>>4), VGPR1 = K+1.
// C/D (16x16 f32, 8 VGPRs): VGPR g -> M = g + 8*(lane>>4), N = lane&15.
// ---------------------------------------------------------------------------
template<int NT, int KDIM, bool BIAS, bool RELU>
__launch_bounds__(256)
__global__ void gemm_wmma_f32(const float* __restrict__ X,
                              const float* __restrict__ W,
                              const float* __restrict__ bias,
                              float* __restrict__ Y,
                              int M) {
    constexpr int N = NT * 16;
    const int wave = (blockIdx.x * blockDim.x + threadIdx.x) >> 5;
    const int lane = threadIdx.x & 31;
    const int row0 = wave * 16;
    if (row0 >= M) return;

    const int r      = lane & 15;        // M row (A) / N col (B,C,D) within tile
    const int khalf  = (lane >> 4) * 2;  // 0 or 2

    v8f acc[NT] = {};

    const float* __restrict__ xrow = X + (size_t)(row0 + r) * KDIM;

#pragma unroll
    for (int k = 0; k < KDIM; k += 4) {
        const int kk = k + khalf;
        // aligned 8-byte load of two consecutive K values for this lane
        v2f a = *(const v2f*)(xrow + kk);
#pragma unroll
        for (int t = 0; t < NT; ++t) {
            const int col = t * 16 + r;
            v2f b;
            b.x = W[(size_t)kk * N + col];
            b.y = W[(size_t)(kk + 1) * N + col];
            acc[t] = __builtin_amdgcn_wmma_f32_16x16x4_f32(
                /*neg_a=*/false, a, /*neg_b=*/false, b,
                /*c_mod=*/(short)0, acc[t],
                /*reuse_a=*/false, /*reuse_b=*/false);
        }
    }

    // Epilogue: scatter 8 accumulator rows per lane
    const int rowoff = (lane >> 4) * 8;
#pragma unroll
    for (int t = 0; t < NT; ++t) {
        const int col = t * 16 + r;
        const float bv = BIAS ? bias[col] : 0.0f;
#pragma unroll
        for (int g = 0; g < 8; ++g) {
            float v = acc[t][g] + bv;
            if (RELU) v = fmaxf(v, 0.0f);
            Y[(size_t)(row0 + rowoff + g) * N + col] = v;
        }
    }
}

// ---------------------------------------------------------------------------
// Edge aggregation: agg[dst] += h[src] * (dis[src]*dis[dst])
// 16 threads per edge, float4 per thread -> 256B coalesced gather per edge,
// native f32 atomics into L2-resident accumulation buffer.
// ---------------------------------------------------------------------------
__global__ void edge_scatter(const float* __restrict__ h,
                             const int* __restrict__ src,
                             const int* __restrict__ dst,
                             const float* __restrict__ dis,
                             float* __restrict__ agg,
                             int ne) {
    const int tid = blockIdx.x * blockDim.x + threadIdx.x;
    const int e = tid >> 4;
    if (e >= ne) return;
    const int f = (tid & 15) * 4;
    const int s = src[e];
    const int d = dst[e];
    const float norm = dis[s] * dis[d];
    const float4 hv = *(const float4*)(h + (size_t)s * 64 + f);
    float* ap = agg + (size_t)d * 64 + f;
    unsafeAtomicAdd(ap + 0, hv.x * norm);
    unsafeAtomicAdd(ap + 1, hv.y * norm);
    unsafeAtomicAdd(ap + 2, hv.z * norm);
    unsafeAtomicAdd(ap + 3, hv.w * norm);
}

// ---------------------------------------------------------------------------
// Self-loop + bias + relu:  out = relu(agg + h*dis^2 + b)
// ---------------------------------------------------------------------------
__global__ void selfloop_bias_relu(const float* __restrict__ agg,
                                   const float* __restrict__ h,
                                   const float* __restrict__ dis,
                                   const float* __restrict__ b,
                                   float* __restrict__ out,
                                   int n) {
    const int tid = blockIdx.x * blockDim.x + threadIdx.x;
    const int i = tid >> 4;
    if (i >= n) return;
    const int f = (tid & 15) * 4;
    const float d2 = dis[i] * dis[i];
    const float4 av = *(const float4*)(agg + (size_t)i * 64 + f);
    const float4 hv = *(const float4*)(h + (size_t)i * 64 + f);
    const float4 bv = *(const float4*)(b + f);
    float4 o;
    o.x = fmaxf(av.x + hv.x * d2 + bv.x, 0.0f);
    o.y = fmaxf(av.y + hv.y * d2 + bv.y, 0.0f);
    o.z = fmaxf(av.z + hv.z * d2 + bv.z, 0.0f);
    o.w = fmaxf(av.w + hv.w * d2 + bv.w, 0.0f);
    *(float4*)(out + (size_t)i * 64 + f) = o;
}

// ---------------------------------------------------------------------------
// Final projection: out[i, 0:2] = hd[i, 0:128] @ Wo[128x2] + bo
// ---------------------------------------------------------------------------
__global__ void out_proj(const float* __restrict__ hd,
                         const float* __restrict__ Wo,
                         const float* __restrict__ bo,
                         float* __restrict__ out,
                         int n) {
    const int i = blockIdx.x * blockDim.x + threadIdx.x;
    if (i >= n) return;
    const float* h = hd + (size_t)i * 128;
    float a0 = bo[0], a1 = bo[1];
#pragma unroll 8
    for (int k = 0; k < 128; ++k) {
        const float hv = h[k];
        a0 += hv * Wo[k * 2 + 0];
        a1 += hv * Wo[k * 2 + 1];
    }
    out[(size_t)i * 2 + 0] = a0;
    out[(size_t)i * 2 + 1] = a1;
}

// ---------------------------------------------------------------------------
// Launch
// ---------------------------------------------------------------------------
extern "C" void kernel_launch(void* const* d_in, const int* in_sizes, int n_in,
                              void* d_out, int out_size, void* d_ws, size_t ws_size,
                              hipStream_t stream) {
    const float* x  = (const float*)d_in[0];
    const int*   ei = (const int*)d_in[1];
    const float* W1 = (const float*)d_in[2];
    const float* b1 = (const float*)d_in[3];
    const float* W2 = (const float*)d_in[4];
    const float* b2 = (const float*)d_in[5];
    const float* Wd = (const float*)d_in[6];
    const float* bd = (const float*)d_in[7];
    const float* Wo = (const float*)d_in[8];
    const float* bo = (const float*)d_in[9];
    float* out = (float*)d_out;

    const int N = in_sizes[0] / 64;   // 100000
    const int E = in_sizes[1] / 2;    // 1600000
    const int* src = ei;
    const int* dst = ei + E;

    // Workspace carve-out (all offsets 16B-aligned: N*4 = 400000 is 16-divisible)
    float* dis  = (float*)d_ws;          // N
    float* bufA = dis  + N;              // N*64  : h (pre-aggregation)
    float* bufB = bufA + (size_t)N * 64; // N*64  : agg accumulator
    float* bufC = bufB + (size_t)N * 64; // N*64  : post-relu conv output
    float* bufD = bufC + (size_t)N * 64; // N*128 : dense hidden
    (void)ws_size; (void)n_in; (void)out_size;

    const int TB = 256;
    const int gemm_waves   = N / 16;                 // 6250 (exact)
    const int gemm_blocks  = (gemm_waves * 32 + TB - 1) / TB;
    const int edge_blocks  = (E * 16 + TB - 1) / TB;
    const int node16_blks  = (N * 16 + TB - 1) / TB;

    // ---- degree / normalization ----
    zero_f32<<<(N + TB - 1) / TB, TB, 0, stream>>>(dis, N);
    deg_count<<<(E + TB - 1) / TB, TB, 0, stream>>>(dst, dis, E);
    finish_dis<<<(N + TB - 1) / TB, TB, 0, stream>>>(dis, N);

    // ---- GCN layer 1 ----
    gemm_wmma_f32<4, 64, false, false><<<gemm_blocks, TB, 0, stream>>>(x, W1, nullptr, bufA, N);
    zero_f32<<<((N * 64) + TB - 1) / TB, TB, 0, stream>>>(bufB, N * 64);
    edge_scatter<<<edge_blocks, TB, 0, stream>>>(bufA, src, dst, dis, bufB, E);
    selfloop_bias_relu<<<node16_blks, TB, 0, stream>>>(bufB, bufA, dis, b1, bufC, N);

    // ---- GCN layer 2 ----
    gemm_wmma_f32<4, 64, false, false><<<gemm_blocks, TB, 0, stream>>>(bufC, W2, nullptr, bufA, N);
    zero_f32<<<((N * 64) + TB - 1) / TB, TB, 0, stream>>>(bufB, N * 64);
    edge_scatter<<<edge_blocks, TB, 0, stream>>>(bufA, src, dst, dis, bufB, E);
    selfloop_bias_relu<<<node16_blks, TB, 0, stream>>>(bufB, bufA, dis, b2, bufC, N);

    // ---- Dense 64 -> 128 with fused bias + relu ----
    gemm_wmma_f32<8, 64, true, true><<<gemm_blocks, TB, 0, stream>>>(bufC, Wd, bd, bufD, N);

    // ---- Output projection 128 -> 2 ----
    out_proj<<<(N + TB - 1) / TB, TB, 0, stream>>>(bufD, Wo, bo, out, N);
}